// H2GCN_4337916969348
// MI455X (gfx1250) — compile-verified
//
#include <hip/hip_runtime.h>

typedef __attribute__((ext_vector_type(2))) float v2f;
typedef __attribute__((ext_vector_type(8))) float v8f;

namespace {
constexpr int IN_C  = 128;
constexpr int HID   = 32;
constexpr int OUT_C = 16;
constexpr int CATD  = 224;   // (2^(K+1)-1)*HID, K=2
}

static __device__ __forceinline__ v8f wmma_f32(v2f a, v2f b, v8f c) {
  // D = A(16x4,f32) * B(4x16,f32) + C(16x16,f32), exact fp32 on the matrix pipe
  return __builtin_amdgcn_wmma_f32_16x16x4_f32(
      /*neg_a=*/false, a, /*neg_b=*/false, b,
      /*c_mod=*/(short)0, c, /*reuse_a=*/false, /*reuse_b=*/false);
}

// ---------------------------------------------------------------------------
// Stage 1: rf[:,0:32] = relu(X @ We).  One wave per 16-row tile; the wave
// computes both 16-wide column tiles of HID=32, sharing its A loads.
// ---------------------------------------------------------------------------
__global__ void __launch_bounds__(32)
h2gcn_embed_wmma(const float* __restrict__ x, const float* __restrict__ w,
                 float* __restrict__ rf)
{
  const int  lane = threadIdx.x;
  const int  m    = lane & 15;       // A row / B-D column index
  const int  half = lane >> 4;       // K sub-slot (A/B) and M-half (D)
  const long row0 = (long)blockIdx.x * 16;

  const float* xr = x + (row0 + m) * IN_C + 2 * half;
  v8f acc0 = {};
  v8f acc1 = {};
  for (int k = 0; k < IN_C; k += 4) {
    v2f a = *(const v2f*)(xr + k);                 // x[row][kk], x[row][kk+1]
    const float* wk = w + (k + 2 * half) * HID + m;
    v2f b0 = { wk[0],      wk[HID]      };         // cols 0..15
    v2f b1 = { wk[16],     wk[HID + 16] };         // cols 16..31
    acc0 = wmma_f32(a, b0, acc0);
    acc1 = wmma_f32(a, b1, acc1);
  }
#pragma unroll
  for (int i = 0; i < 8; ++i) {
    const long r = row0 + i + 8 * half;
    rf[r * CATD + m]      = fmaxf(acc0[i], 0.f);
    rf[r * CATD + 16 + m] = fmaxf(acc1[i], 0.f);
  }
}

// scalar remainder path (only used if n % 16 != 0; n=20000 -> unused)
__global__ void h2gcn_embed_naive(const float* __restrict__ x,
                                  const float* __restrict__ w,
                                  float* __restrict__ rf, int rowStart, int n)
{
  const int r = rowStart + blockIdx.x;
  if (r >= n) return;
  const int t = threadIdx.x;          // 0..HID-1
  float acc = 0.f;
  for (int k = 0; k < IN_C; ++k)
    acc = fmaf(x[(long)r * IN_C + k], w[k * HID + t], acc);
  rf[(long)r * CATD + t] = fmaxf(acc, 0.f);
}

// ---------------------------------------------------------------------------
// SpMM + ReLU: out[u, t] = relu( sum_{e: row[e]==u} val[e] * rin[col[e], t] ).
// Edge lists are row-sorted (built via np.unique on row*n+col), so each
// node's edges are one contiguous range found by binary search -> no atomics.
// blockDim.x == feature width C (32 or 64); lanes own columns -> coalesced
// gathers of rin rows.  rin/out are column-slices of the [N,224] matrix.
// ---------------------------------------------------------------------------
__global__ void h2gcn_spmm_relu(const int* __restrict__ row,
                                const int* __restrict__ col,
                                const float* __restrict__ val, int nE,
                                const float* __restrict__ rin,
                                float* __restrict__ out)
{
  const int u = blockIdx.x;
  const int t = threadIdx.x;

  // lower_bound(row, u)
  int lo = 0, hi = nE;
  while (lo < hi) { int mid = (lo + hi) >> 1; if (row[mid] <  u) lo = mid + 1; else hi = mid; }
  const int beg = lo;
  // upper_bound(row, u)
  hi = nE;
  while (lo < hi) { int mid = (lo + hi) >> 1; if (row[mid] <= u) lo = mid + 1; else hi = mid; }
  const int end = lo;

  float acc = 0.f;
  for (int e = beg; e < end; ++e) {
    if (e + 8 < end)   // hide gather latency: emits global_prefetch_b8
      __builtin_prefetch(rin + (long)col[e + 8] * CATD + t, 0, 0);
    acc = fmaf(val[e], rin[(long)col[e] * CATD + t], acc);
  }
  out[(long)u * CATD + t] = fmaxf(acc, 0.f);
}

// ---------------------------------------------------------------------------
// Stage 3: out = Rfinal[N,224] @ Wc[224,16].  One wave per 16-row tile,
// single 16-wide column tile, 56 k-steps of V_WMMA_F32_16X16X4_F32.
// ---------------------------------------------------------------------------
__global__ void __launch_bounds__(32)
h2gcn_classify_wmma(const float* __restrict__ rf, const float* __restrict__ w,
                    float* __restrict__ out)
{
  const int  lane = threadIdx.x;
  const int  m    = lane & 15;
  const int  half = lane >> 4;
  const long row0 = (long)blockIdx.x * 16;

  const float* ar = rf + (row0 + m) * CATD + 2 * half;
  v8f acc = {};
  for (int k = 0; k < CATD; k += 4) {
    v2f a = *(const v2f*)(ar + k);
    const float* wk = w + (k + 2 * half) * OUT_C + m;
    v2f b = { wk[0], wk[OUT_C] };
    acc = wmma_f32(a, b, acc);
  }
#pragma unroll
  for (int i = 0; i < 8; ++i) {
    const long r = row0 + i + 8 * half;
    out[r * OUT_C + m] = acc[i];
  }
}

__global__ void h2gcn_classify_naive(const float* __restrict__ rf,
                                     const float* __restrict__ w,
                                     float* __restrict__ out, int rowStart, int n)
{
  const int r = rowStart + blockIdx.x;
  if (r >= n) return;
  const int t = threadIdx.x;          // 0..OUT_C-1
  float acc = 0.f;
  for (int k = 0; k < CATD; ++k)
    acc = fmaf(rf[(long)r * CATD + k], w[k * OUT_C + t], acc);
  out[(long)r * OUT_C + t] = acc;
}

// ---------------------------------------------------------------------------
// Input order: x, w_embed, w_classify, a1_row, a1_col, a1_val,
//              a2_row, a2_col, a2_val
// Workspace: rf = [n, 224] fp32 feature matrix (17.9 MB for n=20000);
//   cols [0,32)=r0, [32,64)=A1*r0, [64,96)=A2*r0, [96,160)=A1*r1cat,
//   [160,224)=A2*r1cat.  All slices are fully written before being read;
//   stream order gives the round-1 -> round-2 dependency.
// ---------------------------------------------------------------------------
extern "C" void kernel_launch(void* const* d_in, const int* in_sizes, int n_in,
                              void* d_out, int out_size, void* d_ws, size_t ws_size,
                              hipStream_t stream)
{
  const float* x   = (const float*)d_in[0];
  const float* w_e = (const float*)d_in[1];
  const float* w_c = (const float*)d_in[2];
  const int*   a1r = (const int*)d_in[3];
  const int*   a1c = (const int*)d_in[4];
  const float* a1v = (const float*)d_in[5];
  const int*   a2r = (const int*)d_in[6];
  const int*   a2c = (const int*)d_in[7];
  const float* a2v = (const float*)d_in[8];

  const int n  = in_sizes[0] / IN_C;   // 20000
  const int E1 = in_sizes[3];
  const int E2 = in_sizes[6];

  float* rf = (float*)d_ws;            // n x 224

  const int mt  = n / 16;              // exact for n=20000
  const int rem = n - mt * 16;

  if (mt)  h2gcn_embed_wmma <<<mt, 32, 0, stream>>>(x, w_e, rf);
  if (rem) h2gcn_embed_naive<<<rem, HID, 0, stream>>>(x, w_e, rf, mt * 16, n);

  // round 1: C = 32, input cols [0,32)
  h2gcn_spmm_relu<<<n, 32, 0, stream>>>(a1r, a1c, a1v, E1, rf + 0,  rf + 32);
  h2gcn_spmm_relu<<<n, 32, 0, stream>>>(a2r, a2c, a2v, E2, rf + 0,  rf + 64);
  // round 2: C = 64, input cols [32,96)
  h2gcn_spmm_relu<<<n, 64, 0, stream>>>(a1r, a1c, a1v, E1, rf + 32, rf + 96);
  h2gcn_spmm_relu<<<n, 64, 0, stream>>>(a2r, a2c, a2v, E2, rf + 32, rf + 160);

  if (mt)  h2gcn_classify_wmma <<<mt, 32, 0, stream>>>(rf, w_c, (float*)d_out);
  if (rem) h2gcn_classify_naive<<<rem, OUT_C, 0, stream>>>(rf, w_c, (float*)d_out,
                                                           mt * 16, n);
}